// ZIF_Loglikelihood_18451179504444
// MI455X (gfx1250) — compile-verified
//
#include <hip/hip_runtime.h>
#include <math.h>

typedef __attribute__((ext_vector_type(2))) float v2f;
typedef __attribute__((ext_vector_type(4))) float f4;
typedef __attribute__((ext_vector_type(8))) float v8f;

#define BATCH 256
#define NDIM  4096
#define KDIM  64
#define NB    128     // columns of C staged per LDS block
#define CSTR  132     // padded LDS row stride -> (4*r + c) % 64 banks, conflict-free
#define ASTR  68      // padded stride for 64x64 A in LDS
#define NTHREADS 320  // 10 waves: one per upper-triangle 16x16 tile of symmetric A

__launch_bounds__(NTHREADS, 1)
__global__ void zif_loglik_kernel(const float* __restrict__ targets,
                                  const float* __restrict__ rho,
                                  const float* __restrict__ qs,
                                  const float* __restrict__ means,
                                  const float* __restrict__ Cmat,
                                  const float* __restrict__ psi,
                                  float* __restrict__ out)
{
    __shared__ float Clds[64 * CSTR];    // C block           (8448 floats)
    __shared__ float Cwlds[64 * CSTR];   // w-scaled C block  (8448 floats)
    __shared__ float wlds[NB];
    __shared__ float wdlds[NB];
    __shared__ float Alds[64 * ASTR];    // 64x64 A           (4352 floats)
    __shared__ float ulds[KDIM];

    const int b    = blockIdx.x;
    const int tid  = threadIdx.x;
    const int lane = tid & 31;
    const int wid  = tid >> 5;      // 0..9
    // upper-triangle tile map: wid -> (ti, tj), ti <= tj
    const int ti    = (wid >= 4) + (wid >= 7) + (wid >= 9);
    const int start = (ti == 0) ? 0 : (ti == 1) ? 4 : (ti == 2) ? 7 : 9;
    const int tj    = ti + (wid - start);
    const int l15  = lane & 15;
    const int koff = (lane >= 16) ? 2 : 0;   // K split per ISA 16x4 f32 layout

    const float rho_s  = rho[0];
    const float logrho = logf(rho_s);

    float s_uni = 0.f, s_nb = 0.f, s_mlpsi = 0.f, s_mlq = 0.f, s_negy = 0.f, s_wdd = 0.f;
    float u_part = 0.f;
    v8f acc = {0.f, 0.f, 0.f, 0.f, 0.f, 0.f, 0.f, 0.f};

    const float* trow = targets + (size_t)b * NDIM;
    const float* qrow = qs      + (size_t)b * NDIM;
    const float* mrow = means   + (size_t)b * NDIM;

    for (int n0 = 0; n0 < NDIM; n0 += NB) {
        // ---- phase A: per-element w, w*d, scalar accumulators + C block load ----
        if (tid < NB) {
            int n   = n0 + tid;
            float t = trow[n];
            bool gm = (t >= rho_s);
            float p = psi[n];
            float q = qrow[n];
            float mu = mrow[n];
            float w = 0.f, d = 0.f;
            if (gm) {
                float y = logf(t);
                w = 1.0f / p;
                d = y - mu;
                s_nb    += 1.0f;
                s_mlpsi += logf(p);
                s_mlq   += logf(q);
                s_negy  -= y;             // sum m * logdet_elem (= -y)
            } else {
                s_uni += log1pf(-q) - logrho;
            }
            wlds[tid]  = w;
            wdlds[tid] = w * d;
            s_wdd += w * d * d;
        }
        for (int idx = tid; idx < 64 * (NB / 4); idx += NTHREADS) {  // 2048 float4s
            int r  = idx >> 5;
            int c4 = (idx & 31) * 4;
            *(f4*)(Clds + r * CSTR + c4) =
                *(const f4*)(Cmat + (size_t)r * NDIM + n0 + c4);
            if ((idx & 7) == 0 && (n0 + NB) < NDIM)   // one prefetch per 128B line
                __builtin_prefetch(Cmat + (size_t)r * NDIM + n0 + NB + c4, 0, 1);
        }
        __syncthreads();

        // ---- phase B: Cw = w .* C  and  u partials ----
        for (int e = tid; e < 64 * NB; e += NTHREADS) {
            int r = e >> 7;
            int c = e & (NB - 1);
            Cwlds[r * CSTR + c] = Clds[r * CSTR + c] * wlds[c];
        }
        if (tid < 256) {
            const float* cr = Clds + (tid >> 2) * CSTR + (tid & 3);
            const float* wr = wdlds + (tid & 3);
            #pragma unroll
            for (int i = 0; i < NB / 4; ++i)
                u_part += cr[4 * i] * wr[4 * i];
        }
        __syncthreads();

        // ---- phase C: WMMA  acc += Cw_tilerow * C_tilecol^T ----
        {
            const float* arow = Cwlds + (ti * 16 + l15) * CSTR + koff;
            const float* brow = Clds  + (tj * 16 + l15) * CSTR + koff;
            #pragma unroll 4
            for (int kk = 0; kk < NB; kk += 4) {
                v2f av = *(const v2f*)(arow + kk);   // ds_load_b64, 8B aligned
                v2f bv = *(const v2f*)(brow + kk);
                acc = __builtin_amdgcn_wmma_f32_16x16x4_f32(
                          false, av, false, bv, (short)0, acc, false, false);
            }
        }
        __syncthreads();   // protect LDS blocks for next iteration
    }

    // ---- spill A tiles (and symmetric mirror) to LDS ----
    {
        int rbase = ti * 16 + ((lane >= 16) ? 8 : 0);
        int col   = tj * 16 + l15;
        #pragma unroll
        for (int v = 0; v < 8; ++v) {
            Alds[(rbase + v) * ASTR + col] = acc[v];
            if (ti != tj)
                Alds[col * ASTR + (rbase + v)] = acc[v];   // A is symmetric
        }
    }

    // ---- reductions in Clds scratch ----
    float* red = Clds;            // u partials [0..319]; only 0..255 meaningful
    float* r1 = Clds + 512;
    float* r2 = Clds + 1024;
    float* r3 = Clds + 1536;
    float* r4 = Clds + 2048;
    float* r5 = Clds + 2560;
    float* r6 = Clds + 3072;
    red[tid] = u_part;
    r1[tid] = s_uni;  r2[tid] = s_nb;   r3[tid] = s_mlpsi;
    r4[tid] = s_mlq;  r5[tid] = s_negy; r6[tid] = s_wdd;
    __syncthreads();

    if (tid < KDIM) {
        float s = 0.f;
        #pragma unroll
        for (int i = 0; i < 4; ++i) s += red[tid * 4 + i];
        ulds[tid] = s;
        Alds[tid * ASTR + tid] += 1.0f;   // A = I + C W C^T
    }
    if (tid < 64) {   // fold 320 -> 256
        r1[tid] += r1[tid + 256]; r2[tid] += r2[tid + 256];
        r3[tid] += r3[tid + 256]; r4[tid] += r4[tid + 256];
        r5[tid] += r5[tid + 256]; r6[tid] += r6[tid + 256];
    }
    __syncthreads();
    for (int off = 128; off > 0; off >>= 1) {
        if (tid < off) {
            r1[tid] += r1[tid + off]; r2[tid] += r2[tid + off];
            r3[tid] += r3[tid + off]; r4[tid] += r4[tid + off];
            r5[tid] += r5[tid + off]; r6[tid] += r6[tid + off];
        }
        __syncthreads();
    }

    // ---- cooperative 64x64 Cholesky (in place, full-square trailing update) ----
    for (int k = 0; k < KDIM; ++k) {
        if (tid == 0) Alds[k * ASTR + k] = sqrtf(Alds[k * ASTR + k]);
        __syncthreads();
        float invd = 1.0f / Alds[k * ASTR + k];
        int rem = KDIM - 1 - k;
        if (tid < rem) {
            int i = k + 1 + tid;
            Alds[i * ASTR + k] *= invd;
        }
        __syncthreads();
        int tot = rem * rem;
        for (int idx = tid; idx < tot; idx += NTHREADS) {
            int i = k + 1 + idx / rem;
            int j = k + 1 + idx % rem;
            Alds[i * ASTR + j] -= Alds[i * ASTR + k] * Alds[j * ASTR + k];
        }
        __syncthreads();
    }

    // ---- forward solve L y = u  (then u^T A^-1 u = ||y||^2) ----
    for (int k = 0; k < KDIM; ++k) {
        if (tid == 0) ulds[k] /= Alds[k * ASTR + k];
        __syncthreads();
        if (tid > k && tid < KDIM) ulds[tid] -= Alds[tid * ASTR + k] * ulds[k];
        __syncthreads();
    }

    // ---- final combine ----
    if (tid == 0) {
        float logdetA = 0.f, uAu = 0.f;
        for (int k = 0; k < KDIM; ++k) {
            logdetA += logf(Alds[k * ASTR + k]);
            uAu     += ulds[k] * ulds[k];
        }
        logdetA *= 2.0f;
        const float LOG2PI = 1.8378770664093453f;
        float t_uni = r1[0], t_nb = r2[0], t_mlpsi = r3[0];
        float t_mlq = r4[0], t_negy = r5[0], t_wdd = r6[0];
        float logdet_cov = t_mlpsi + logdetA;
        float quad  = t_wdd - uAu;
        float gauss = -0.5f * (t_nb * LOG2PI + logdet_cov + quad);
        out[b] = t_uni + gauss + t_negy + t_mlq;
    }
}

extern "C" void kernel_launch(void* const* d_in, const int* in_sizes, int n_in,
                              void* d_out, int out_size, void* d_ws, size_t ws_size,
                              hipStream_t stream) {
    (void)in_sizes; (void)n_in; (void)d_ws; (void)ws_size; (void)out_size;
    const float* targets = (const float*)d_in[0];
    const float* rho     = (const float*)d_in[1];
    const float* qs      = (const float*)d_in[2];
    const float* means   = (const float*)d_in[3];
    const float* Cmat    = (const float*)d_in[4];
    const float* psi     = (const float*)d_in[5];
    float* out = (float*)d_out;
    zif_loglik_kernel<<<BATCH, NTHREADS, 0, stream>>>(targets, rho, qs, means, Cmat, psi, out);
}